// ham_func_3body_54099408060844
// MI455X (gfx1250) — compile-verified
//
#include <hip/hip_runtime.h>
#include <cstdint>

// 3-body Hamiltonian derivative: elementwise over B rows of 12 f32.
// Memory-bound: 192 MB traffic -> ~8.2us at 23.3 TB/s. Strategy:
//   - async global->LDS staging (gfx1250 GLOBAL_LOAD_ASYNC_TO_LDS_B64),
//     lane-consecutive 8B chunks => perfectly coalesced HBM streams
//   - LDS rows padded 48B->56B (14 words): 14*(i-j)%64==0 iff i==j (mod 32)
//     => bank-conflict-free wave32 access, 8B alignment preserved
//   - compute in registers, write back to same LDS row, async LDS->global out
//   - ordering via s_wait_asynccnt / s_wait_dscnt split counters + barrier

#define TPB 256
#define ROW_WORDS 14            // 12 payload floats + 2 pad words = 56 bytes
#define ROW_BYTES (ROW_WORDS * 4)

__global__ __launch_bounds__(TPB) void body3_deriv_kernel(
    const float* __restrict__ y, const float* __restrict__ M,
    float* __restrict__ out, int B)
{
    __shared__ float smem[TPB * ROW_WORDS];      // 14336 B per block

    const int tid  = threadIdx.x;
    const int base = blockIdx.x * TPB;                      // first sample
    const int n    = (B - base < TPB) ? (B - base) : TPB;   // samples here
    const int nch  = n * 6;                                 // 8B chunks

    // LDS offset of smem (low 32 bits of the generic pointer are the
    // wave-relative LDS byte address on gfx1250).
    const uint32_t lds0 = (uint32_t)(uintptr_t)&smem[0];

    // ---- coalesced async copy: global -> LDS (padded rows) ----
    // chunk t covers sample t/6, float-pair t%6. Consecutive lanes touch
    // consecutive 8B global addresses; each lane scatters into its padded row.
    #pragma unroll
    for (int k = 0; k < 6; ++k) {
        int t = k * TPB + tid;
        if (t < nch) {
            int s = t / 6;
            int p = t - s * 6;
            uint32_t lds_addr = lds0 + (uint32_t)(s * ROW_BYTES + p * 8);
            uint32_t voff     = (uint32_t)(base * 48 + t * 8);
            asm volatile("global_load_async_to_lds_b64 %0, %1, %2"
                         :: "v"(lds_addr), "v"(voff), "s"(y) : "memory");
        }
    }
    asm volatile("s_wait_asynccnt 0" ::: "memory");  // our copies landed
    __syncthreads();                                  // everyone's copies visible

    // ---- per-sample compute (registers only) ----
    if (tid < n) {
        float* row = &smem[tid * ROW_WORDS];
        float x1 = row[0],  y1 = row[1];
        float x2 = row[2],  y2 = row[3];
        float x3 = row[4],  y3 = row[5];
        float p1x = row[6], p1y = row[7];
        float p2x = row[8], p2y = row[9];
        float p3x = row[10], p3y = row[11];

        float M0 = M[0], M1 = M[1], M2 = M[2];

        float dx = x1 - x2, dy = y1 - y2;
        float d2 = dx * dx + dy * dy;
        float r  = sqrtf(d2);
        float a  = (M0 * M1) / (d2 * r);     // G*M0*M1 / r^3, G = 1

        dx = x1 - x3; dy = y1 - y3;
        d2 = dx * dx + dy * dy;
        r  = sqrtf(d2);
        float b = (M0 * M2) / (d2 * r);

        dx = x2 - x3; dy = y2 - y3;
        d2 = dx * dx + dy * dy;
        r  = sqrtf(d2);
        float c = (M1 * M2) / (d2 * r);

        // dq = p / m
        row[0] = p1x / M0;
        row[1] = p1y / M0;
        row[2] = p2x / M1;
        row[3] = p2y / M1;
        row[4] = p3x / M2;
        row[5] = p3y / M2;
        // dp
        row[6]  = -(a + b) * x1 + a * x2 + b * x3;
        row[7]  = -(a + b) * y1 + a * y2 + b * y3;
        row[8]  =  a * x1 - (a + c) * x2 + c * x3;
        row[9]  =  a * y1 - (a + c) * y2 + c * y3;
        row[10] =  b * x1 + c * x2 - (b + c) * x3;
        row[11] =  b * y1 + c * y2 - (b + c) * y3;
    }

    // make this wave's LDS writes visible to the async-DMA reader
    asm volatile("s_wait_dscnt 0" ::: "memory");

    // ---- coalesced async store: LDS -> global ----
    // each wave drains only its own 32 rows (written by its own lanes),
    // so no extra barrier is required. chunk addresses are lane-consecutive.
    const int lane  = tid & 31;
    const int wbase = tid & ~31;           // wave's first block-local sample
    #pragma unroll
    for (int j = 0; j < 6; ++j) {
        int u = j * 32 + lane;             // wave-local chunk 0..191
        int g = wbase * 6 + u;             // block-local chunk
        if (g < nch) {
            int s = wbase + u / 6;
            int p = u - (u / 6) * 6;
            uint32_t lds_addr = lds0 + (uint32_t)(s * ROW_BYTES + p * 8);
            uint32_t voff     = (uint32_t)(base * 48 + g * 8);
            asm volatile("global_store_async_from_lds_b64 %0, %1, %2"
                         :: "v"(voff), "v"(lds_addr), "s"(out) : "memory");
        }
    }
    // S_ENDPGM performs an implicit wait-idle, so the trailing async stores
    // are guaranteed complete before the wave retires.
}

extern "C" void kernel_launch(void* const* d_in, const int* in_sizes, int n_in,
                              void* d_out, int out_size, void* d_ws, size_t ws_size,
                              hipStream_t stream) {
    (void)n_in; (void)d_ws; (void)ws_size;
    const float* y = (const float*)d_in[0];   // [B,12] f32
    const float* M = (const float*)d_in[1];   // [3] f32
    // d_in[2] = t, unused
    float* out = (float*)d_out;               // [B,12] f32
    const int B = in_sizes[0] / 12;
    const int blocks = (B + TPB - 1) / TPB;
    hipLaunchKernelGGL(body3_deriv_kernel, dim3(blocks), dim3(TPB), 0, stream,
                       y, M, out, B);
}